// ExponentialSmoothingNet_31696858644986
// MI455X (gfx1250) — compile-verified
//
#include <hip/hip_runtime.h>

typedef __attribute__((ext_vector_type(2))) float v2f;
typedef __attribute__((ext_vector_type(8))) float v8f;

// Problem shape (fixed by the reference).
constexpr int T = 2048;
constexpr int F = 512;
constexpr int BATCH = 64;
// EMA with alpha=0.5: weight of x[t] in the final state is 2^-(T-t).
// fp32 cannot represent anything below 2^-149, so timesteps with lag > 148
// contribute exactly 0.0f. Process the last 160 steps (multiple of K=4);
// the few extra chunks have weight == 0.0f and add nothing.
constexpr int KPROC = 160;

// One wave32 per 16-feature output tile of one batch.
// C[16x16] += A[16x4] x B[4x16] with
//   A[m,k] = x[b, t0+k, f0+m]          (16 consecutive features, 4 timesteps)
//   B[k,n] = 2^-(T-(t0+k))  for all n  (weight broadcast across columns)
// so every C column equals y[b, f0+0..15].
__global__ __launch_bounds__(256) void ExponentialSmoothingNet_ema_wmma(
    const float* __restrict__ x, float* __restrict__ out) {
  const int lane = threadIdx.x & 31;
  const int wave = threadIdx.x >> 5;
  const int tile = blockIdx.x * 8 + wave;   // 2048 tiles = 64 batches * 32 f-tiles
  const int b    = tile >> 5;
  const int f0   = (tile & 31) << 4;

  // ISA layout for 32-bit 16x4 A fragment (wave32):
  //   lanes 0-15 : M = lane,    V0 = K0, V1 = K1
  //   lanes 16-31: M = lane-16, V0 = K2, V1 = K3
  // B 4x16 mirrors this split: lanes 0-15 carry K0/K1, lanes 16-31 carry K2/K3.
  const int m     = lane & 15;
  const int kbase = (lane >> 4) << 1;       // 0 or 2

  const float* xb = x + (size_t)b * T * F + (size_t)f0 + m;

  v8f c = {};  // accumulate oldest (smallest weights) -> newest for accuracy
#pragma unroll 4
  for (int t0 = T - KPROC; t0 < T; t0 += 4) {
    v2f a, w;
    a.x = xb[(size_t)(t0 + kbase) * F];
    a.y = xb[(size_t)(t0 + kbase + 1) * F];
    // Exact power-of-two weights; ldexpf handles the denormal tail correctly.
    w.x = ldexpf(1.0f, (t0 + kbase) - T);
    w.y = ldexpf(1.0f, (t0 + kbase + 1) - T);
    // 8 args: (neg_a, A, neg_b, B, c_mod, C, reuse_a, reuse_b)
    c = __builtin_amdgcn_wmma_f32_16x16x4_f32(
        /*neg_a=*/false, a, /*neg_b=*/false, w,
        /*c_mod=*/(short)0, c, /*reuse_a=*/false, /*reuse_b=*/false);
  }

  // C layout: VGPR r holds M=r (lanes 0-15) and M=8+r (lanes 16-31); every
  // column N is identical, so lane 0 writes features f0+0..7 and lane 16
  // writes f0+8..15. Output is [B, 1, F] flattened.
  if (m == 0) {
    float* o = out + (size_t)b * F + f0 + ((lane >> 4) << 3);
#pragma unroll
    for (int i = 0; i < 8; ++i) o[i] = c[i];
  }
}

extern "C" void kernel_launch(void* const* d_in, const int* in_sizes, int n_in,
                              void* d_out, int out_size, void* d_ws, size_t ws_size,
                              hipStream_t stream) {
  const float* x = (const float*)d_in[0];
  float* out = (float*)d_out;
  (void)in_sizes; (void)n_in; (void)out_size; (void)d_ws; (void)ws_size;

  // 2048 tiles, 8 waves (256 threads) per block -> 256 blocks.
  dim3 grid(256, 1, 1), block(256, 1, 1);
  hipLaunchKernelGGL(ExponentialSmoothingNet_ema_wmma, grid, block, 0, stream,
                     x, out);
}